// SparseMLP_74629351735474
// MI455X (gfx1250) — compile-verified
//
#include <hip/hip_runtime.h>
#include <hip/hip_bf16.h>
#include <stdint.h>

// ---------------------------------------------------------------------------
// SparseMLP on MI455X (gfx1250), fp32 end-to-end.
//
// out[nz] = (x[nz] @ W1) @ W2 ; other row-blocks zero.
//
// Memory-bound problem (~1.1 TFLOP over ~1.2 GB min HBM traffic => ~52us at
// 23.3 TB/s; compute need only exceed ~21 TFLOPS), so we keep full fp32
// precision and use the CDNA5 fp32 tensor op V_WMMA_F32_16X16X4_F32.
//
// CDNA5 data paths used:
//  - v_wmma_f32_16x16x4_f32 (VOP3P matrix pipe), fragments loaded as single
//    ds_load_b64 into even-aligned VGPR pairs (K,K+1 contiguous in LDS).
//  - global_load_async_to_lds_b128 (ASYNCcnt-tracked DMA, no VGPR staging)
//    for the A tile, fenced with s_wait_asynccnt before the barrier.
//  - global_prefetch_b8 (via __builtin_prefetch) for the next B tile.
// ---------------------------------------------------------------------------

typedef float v2f __attribute__((ext_vector_type(2)));
typedef float v8f __attribute__((ext_vector_type(8)));

#define TILE_M 128
#define TILE_N 128
#define TILE_K 16
#define A_PAD 4   // As row stride 20 dwords = 80B: 16B aligned, 20*m%64 distinct
#define B_PAD 2   // Bs row stride 18 dwords = 72B:  8B aligned, 18*n%64 distinct

// Low 32 bits of a generic pointer into LDS == LDS byte address (ISA 10.2).
__device__ __forceinline__ uint32_t lds_addr32(const void* p) {
    return (uint32_t)(uintptr_t)p;
}

// Generic 128x128-tile fp32 WMMA GEMM:  C = A(128 rows) * B,  K-contraction.
// gatherA: A row-block chosen via nz[blockIdx.y]; else dense blockIdx.y.
// scatterC: C row-block chosen via nz[blockIdx.y]; else dense blockIdx.y.
__global__ __launch_bounds__(256) void sparse_mlp_gemm_f32_wmma(
    const float* __restrict__ A, int lda,
    const float* __restrict__ B, int ldb,
    float* __restrict__ C, int ldc,
    int K,
    const int* __restrict__ nz,
    int gatherA, int scatterC)
{
    __shared__ float As[TILE_M][TILE_K + A_PAD]; // A tile, memory layout: As[m][k]
    __shared__ float Bs[TILE_N][TILE_K + B_PAD]; // B tile, transposed:    Bs[n][k]

    const int blk   = blockIdx.y;
    const int arow0 = (gatherA  ? nz[blk] : blk) * TILE_M;
    const int crow0 = (scatterC ? nz[blk] : blk) * TILE_M;

    const float* Ab = A + (long long)arow0 * lda;
    const float* Bb = B + (long long)blockIdx.x * TILE_N;
    float*       Cb = C + (long long)crow0 * ldc + (long long)blockIdx.x * TILE_N;

    const int tid  = threadIdx.x;
    const int lane = tid & 31;            // wave32
    const int wave = tid >> 5;            // 0..7
    const int half = lane >> 4;           // 0: lanes 0-15, 1: lanes 16-31
    const int lsub = lane & 15;           // position within half-wave

    // Wave tiling: 4 waves along M (32 rows each), 2 along N (64 cols each).
    const int mbase = (wave & 3) * 32;    // 0,32,64,96
    const int nbase = (wave >> 2) * 64;   // 0,64

    // Per-thread staging assignments.
    const int a_m  = tid >> 1;            // 0..127  (A: 128 rows x 16 floats)
    const int a_f0 = (tid & 1) * 8;       // 0 or 8  (two b128 quads per thread)
    const int b_k  = tid >> 4;            // 0..15   (B: 16 rows x 128 floats)
    const int b_n  = (tid & 15) * 8;      // 0..120

    // LDS destinations for the async A copies (loop-invariant).
    const uint32_t a_lds0 = lds_addr32(&As[a_m][a_f0]);
    const uint32_t a_lds1 = lds_addr32(&As[a_m][a_f0 + 4]);

    v8f acc[2][4] = {};                   // 2 M-subtiles x 4 N-subtiles, 16x16 each

    for (int k0 = 0; k0 < K; k0 += TILE_K) {
        // ---- stage A tile via async DMA to LDS (no transpose needed) ----
        {
            const float* src = Ab + (long long)a_m * lda + k0 + a_f0;
            asm volatile("global_load_async_to_lds_b128 %0, %1, off"
                         :: "v"(a_lds0), "v"(src) : "memory");
            asm volatile("global_load_async_to_lds_b128 %0, %1, off"
                         :: "v"(a_lds1), "v"(src + 4) : "memory");
        }
        // ---- stage B tile through VGPRs with transpose: Bs[n][k] ----
        {
            const float* src = Bb + (long long)(k0 + b_k) * ldb + b_n;
            float4 q0 = *(const float4*)(src);
            float4 q1 = *(const float4*)(src + 4);
            Bs[b_n + 0][b_k] = q0.x;  Bs[b_n + 1][b_k] = q0.y;
            Bs[b_n + 2][b_k] = q0.z;  Bs[b_n + 3][b_k] = q0.w;
            Bs[b_n + 4][b_k] = q1.x;  Bs[b_n + 5][b_k] = q1.y;
            Bs[b_n + 6][b_k] = q1.z;  Bs[b_n + 7][b_k] = q1.w;
        }
        // Prefetch next B tile rows into cache (global_prefetch_b8).
        if (k0 + TILE_K < K) {
            __builtin_prefetch(Bb + (long long)(k0 + TILE_K + b_k) * ldb + b_n, 0, 0);
        }
        // Async copies must have landed in LDS before anyone reads the tile.
        asm volatile("s_wait_asynccnt 0x0" ::: "memory");
        __syncthreads();

        // ---- compute: 4 WMMA K-steps of 4 over the 16-deep tile ----
#pragma unroll
        for (int kk = 0; kk < TILE_K; kk += 4) {
            const int kA = kk + 2 * half;  // lanes 0-15: K=kk+{0,1}; 16-31: kk+{2,3}

            v2f afrag[2];                  // single ds_load_b64 each (K,K+1 adjacent)
#pragma unroll
            for (int i = 0; i < 2; ++i)
                afrag[i] = *(const v2f*)&As[mbase + i * 16 + lsub][kA];

            v2f bfrag[4];
#pragma unroll
            for (int j = 0; j < 4; ++j)
                bfrag[j] = *(const v2f*)&Bs[nbase + j * 16 + lsub][kA];

#pragma unroll
            for (int i = 0; i < 2; ++i)
#pragma unroll
                for (int j = 0; j < 4; ++j)
                    acc[i][j] = __builtin_amdgcn_wmma_f32_16x16x4_f32(
                        /*neg_a=*/false, afrag[i],
                        /*neg_b=*/false, bfrag[j],
                        /*c_mod=*/(short)0, acc[i][j],
                        /*reuse_a=*/false, /*reuse_b=*/false);
        }
        __syncthreads();
    }

    // ---- epilogue: C/D layout -> lanes 0-15 hold M=r, lanes 16-31 hold M=8+r ----
#pragma unroll
    for (int i = 0; i < 2; ++i) {
#pragma unroll
        for (int j = 0; j < 4; ++j) {
            const int rbase = mbase + i * 16 + half * 8;
            const int col   = nbase + j * 16 + lsub;
#pragma unroll
            for (int r = 0; r < 8; ++r) {
                Cb[(long long)(rbase + r) * ldc + col] = acc[i][j][r];
            }
        }
    }
}

// Zero the full output (untouched blocks must be 0; d_out is poisoned once).
__global__ void sparse_mlp_zero_out(float4* __restrict__ p, long long n4)
{
    long long i = (long long)blockIdx.x * blockDim.x + threadIdx.x;
    if (i < n4) p[i] = make_float4(0.f, 0.f, 0.f, 0.f);
}

extern "C" void kernel_launch(void* const* d_in, const int* in_sizes, int n_in,
                              void* d_out, int out_size, void* d_ws, size_t ws_size,
                              hipStream_t stream)
{
    const float* x  = (const float*)d_in[0];   // (8192, 4096) fp32
    const float* w1 = (const float*)d_in[1];   // (4096, 16384) fp32
    const float* w2 = (const float*)d_in[2];   // (16384, 4096) fp32
    const int*   nz = (const int*)d_in[3];     // (num_nz,) int32
    const int num_nz = in_sizes[3];            // == NUM_NZ_BLOCKS (32)

    const int embed  = 4096;
    const int hidden = 16384;
    const int outdim = 4096;

    float* out = (float*)d_out;
    float* h   = (float*)d_ws;                 // num_nz*128*hidden fp32 (256 MB)

    // 1) zero the whole output
    {
        long long n4 = (long long)out_size / 4;
        int blocks = (int)((n4 + 255) / 256);
        sparse_mlp_zero_out<<<blocks, 256, 0, stream>>>((float4*)out, n4);
    }
    // 2) h = gather(x)[nz] @ W1
    {
        dim3 grid(hidden / TILE_N, num_nz);
        sparse_mlp_gemm_f32_wmma<<<grid, 256, 0, stream>>>(
            x, embed, w1, hidden, h, hidden, embed, nz, /*gatherA=*/1, /*scatterC=*/0);
    }
    // 3) out[nz] = h @ W2
    {
        dim3 grid(outdim / TILE_N, num_nz);
        sparse_mlp_gemm_f32_wmma<<<grid, 256, 0, stream>>>(
            h, hidden, w2, outdim, out, outdim, hidden, nz, /*gatherA=*/0, /*scatterC=*/1);
    }
}